// GE2ELoss_82832739271155
// MI455X (gfx1250) — compile-verified
//
#include <hip/hip_runtime.h>
#include <hip/hip_bf16.h>
#include <math.h>

typedef __attribute__((ext_vector_type(2))) float v2f;
typedef __attribute__((ext_vector_type(8))) float v8f;

#define GN 2048   // speakers
#define GM 16     // utterances per speaker
#define GD 512    // embedding dim
#define EPSC 1e-6f

// ---------------------------------------------------------------------------
// Kernel A: per-speaker prepass.
//  - centroids[j][d] = mean_i dvecs[j][i][d]     (to ws, row-major [N][D])
//  - inv_c[j] = 1/||centroid_j||
//  - inv_e[j*16+i] = 1/||dvecs[j][i]||
//  - sim_loo[j*16+i] = clip(cos_loo, eps)*w + b  (leave-one-out, analytic)
// ---------------------------------------------------------------------------
__global__ __launch_bounds__(256) void ge2e_prepass(
    const float* __restrict__ dvecs, const float* __restrict__ wp,
    const float* __restrict__ bp, float* __restrict__ centroids,
    float* __restrict__ inv_e, float* __restrict__ sim_loo,
    float* __restrict__ inv_c) {
  const int j = blockIdx.x;
  const int tid = threadIdx.x;
  const int lane = tid & 31;
  const int wave = tid >> 5;

  __shared__ float Al[GM * GD];  // dvecs[j], row-major [i][d]
  __shared__ float ssum[GD];     // speaker sum over i
  __shared__ float red[8];
  __shared__ float ss_sh;

  const float* src = dvecs + (size_t)j * GM * GD;
#pragma unroll
  for (int s = 0; s < (GM * GD) / 256; ++s) {
    int idx = tid + 256 * s;
    Al[idx] = src[idx];
  }
  __syncthreads();

  // speaker sum + centroid
  for (int d = tid; d < GD; d += 256) {
    float acc = 0.f;
#pragma unroll
    for (int i = 0; i < GM; ++i) acc += Al[i * GD + d];
    ssum[d] = acc;
    centroids[(size_t)j * GD + d] = acc * (1.0f / GM);
  }
  __syncthreads();

  // ||ssum||^2 block reduction
  float p = 0.f;
  for (int d = tid; d < GD; d += 256) p += ssum[d] * ssum[d];
#pragma unroll
  for (int off = 16; off >= 1; off >>= 1) p += __shfl_xor(p, off, 32);
  if (lane == 0) red[wave] = p;
  __syncthreads();
  if (tid == 0) {
    float ss = 0.f;
#pragma unroll
    for (int wv = 0; wv < 8; ++wv) ss += red[wv];
    ss_sh = ss;
    inv_c[j] = (float)GM * rsqrtf(ss);  // c_norm = sqrt(ss)/M
  }
  __syncthreads();
  const float ss = ss_sh;
  const float w = *wp, b = *bp;

  // each wave handles rows {wave, wave+8}
#pragma unroll
  for (int rr = 0; rr < 2; ++rr) {
    int i = wave + 8 * rr;
    float dd = 0.f, ds = 0.f;
#pragma unroll
    for (int s = 0; s < GD / 32; ++s) {
      int d = lane + 32 * s;
      float x = Al[i * GD + d];
      dd += x * x;
      ds += x * ssum[d];
    }
#pragma unroll
    for (int off = 16; off >= 1; off >>= 1) {
      dd += __shfl_xor(dd, off, 32);
      ds += __shfl_xor(ds, off, 32);
    }
    if (lane == 0) {
      float e = sqrtf(dd);
      inv_e[j * GM + i] = 1.0f / e;
      float loo2 = ss - 2.0f * ds + dd;               // ||ssum - d||^2
      float cos_loo = (ds - dd) / (e * sqrtf(loo2));  // (M-1) cancels
      sim_loo[j * GM + i] = fmaxf(cos_loo, EPSC) * w + b;
    }
  }
}

// ---------------------------------------------------------------------------
// Kernel B: fused 16x2048 GEMM (K=512, fp32 WMMA 16x16x4) + softmax sum.
// Since cos is clipped to [EPS, 1], sim <= C := max(w, w*EPS) + b, so we can
// use a FIXED logsumexp shift C: s = sum(exp(sim - C)), lse = C + log(s).
// One WG (8 waves) per speaker j.
// ---------------------------------------------------------------------------
__global__ __launch_bounds__(256) void ge2e_main(
    const float* __restrict__ dvecs, const float* __restrict__ centroids,
    const float* __restrict__ inv_e, const float* __restrict__ sim_loo,
    const float* __restrict__ inv_c, const float* __restrict__ wp,
    const float* __restrict__ bp, float* __restrict__ partials) {
  const int j = blockIdx.x;
  const int tid = threadIdx.x;
  const int lane = tid & 31;
  const int wave = tid >> 5;
  const int half = lane >> 4;  // 0: K={0,1}, 1: K={2,3}
  const int l16 = lane & 15;

  __shared__ float Alds[GD * GM];  // k-major: Alds[k*16 + row]
  __shared__ float ssl[8][GM];     // per-wave partial sumexp per row

  // stage A = dvecs[j] (16x512), transposing to k-major for WMMA A frags
  const float* src = dvecs + (size_t)j * GM * GD;
#pragma unroll
  for (int s = 0; s < (GM * GD) / 256; ++s) {
    int idx = tid + 256 * s;  // idx = i*512 + d  (coalesced global read)
    int i = idx >> 9;
    int d = idx & (GD - 1);
    Alds[d * GM + i] = src[idx];
  }
  __syncthreads();

  const float w = *wp, b = *bp;
  const float C = fmaxf(w, w * EPSC) + b;  // upper bound of sim
  // rows this lane carries in C vgprs: row = r + 8*half
  float ie[8], slr[8];
#pragma unroll
  for (int r = 0; r < 8; ++r) {
    ie[r] = inv_e[j * GM + r + 8 * half];
    slr[r] = sim_loo[j * GM + r + 8 * half];
  }

  float srun[8];
#pragma unroll
  for (int r = 0; r < 8; ++r) srun[r] = 0.f;

  // 128 column tiles of 16 centroids; wave owns tiles wave, wave+8, ...
  for (int ct = wave; ct < GN / 16; ct += 8) {
    const int colbase = ct * 16;
    const int col = colbase + l16;
    // B fragment source: column `col`'s centroid, K split across lane halves
    const float* Bcol = centroids + (size_t)col * GD + 2 * half;

    v8f c = {};
#pragma unroll 8
    for (int kk = 0; kk < GD; kk += 4) {
      v2f a;
      a.x = Alds[(kk + 2 * half) * GM + l16];
      a.y = Alds[(kk + 2 * half + 1) * GM + l16];
      v2f bfrag = *(const v2f*)(Bcol + kk);
      c = __builtin_amdgcn_wmma_f32_16x16x4_f32(
          false, a, false, bfrag, (short)0, c, false, false);
    }

    const float ic = inv_c[col];
    const bool diag = (col == j);
#pragma unroll
    for (int r = 0; r < 8; ++r) {
      float cosv = c[r] * ie[r] * ic;
      float simv = fmaxf(cosv, EPSC) * w + b;  // always computed
      simv = diag ? slr[r] : simv;             // v_cndmask, no branch
      srun[r] += __expf(simv - C);
    }
  }

  // sum partial sumexp across the 16 lanes sharing each row (within halves)
#pragma unroll
  for (int r = 0; r < 8; ++r) {
    float s = srun[r];
#pragma unroll
    for (int off = 8; off >= 1; off >>= 1) s += __shfl_xor(s, off, 32);
    if (l16 == 0) ssl[wave][r + 8 * half] = s;
  }
  __syncthreads();

  // wave 0, lanes 0..15: merge 8 waves per row, compute loss row terms
  if (tid < 16) {
    int row = tid;
    float s = 0.f;
#pragma unroll
    for (int wv = 0; wv < 8; ++wv) s += ssl[wv][row];
    float L = (C + __logf(s)) - sim_loo[j * GM + row];
#pragma unroll
    for (int off = 8; off >= 1; off >>= 1) L += __shfl_xor(L, off, 32);
    if (tid == 0) partials[j] = L;
  }
}

// ---------------------------------------------------------------------------
// Kernel C: sum 2048 per-speaker partials -> d_out[0]
// ---------------------------------------------------------------------------
__global__ __launch_bounds__(256) void ge2e_finish(
    const float* __restrict__ partials, float* __restrict__ out) {
  __shared__ float red[8];
  const int tid = threadIdx.x;
  float p = 0.f;
  for (int i = tid; i < GN; i += 256) p += partials[i];
#pragma unroll
  for (int off = 16; off >= 1; off >>= 1) p += __shfl_xor(p, off, 32);
  if ((tid & 31) == 0) red[tid >> 5] = p;
  __syncthreads();
  if (tid == 0) {
    float s = 0.f;
#pragma unroll
    for (int wv = 0; wv < 8; ++wv) s += red[wv];
    out[0] = s;
  }
}

extern "C" void kernel_launch(void* const* d_in, const int* in_sizes, int n_in,
                              void* d_out, int out_size, void* d_ws, size_t ws_size,
                              hipStream_t stream) {
  const float* dvecs = (const float*)d_in[0];
  const float* wp = (const float*)d_in[1];
  const float* bp = (const float*)d_in[2];
  float* out = (float*)d_out;

  float* centroids = (float*)d_ws;                 // N*D
  float* inv_e     = centroids + (size_t)GN * GD;  // N*M
  float* sim_loo   = inv_e + GN * GM;              // N*M
  float* inv_c     = sim_loo + GN * GM;            // N
  float* partials  = inv_c + GN;                   // N

  ge2e_prepass<<<GN, 256, 0, stream>>>(dvecs, wp, bp, centroids, inv_e,
                                       sim_loo, inv_c);
  ge2e_main<<<GN, 256, 0, stream>>>(dvecs, centroids, inv_e, sim_loo, inv_c,
                                    wp, bp, partials);
  ge2e_finish<<<1, 256, 0, stream>>>(partials, out);
}